// LG_MGC_2731599200834
// MI455X (gfx1250) — compile-verified
//
#include <hip/hip_runtime.h>
#include <hip/hip_bf16.h>
#include <float.h>

typedef float v2f __attribute__((ext_vector_type(2)));
typedef float v8f __attribute__((ext_vector_type(8)));

#define BATCH   256
#define LTOK    576
#define DDIM    1024
#define NTILES  (LTOK / 16)
#define THREADS 256
#define NWAVES  (THREADS / 32)
#define MAXK    64
#define EPS_    1e-8f

__global__ __launch_bounds__(THREADS)
void topk_neg_cos_mean_kernel(const float* __restrict__ iF,
                              const float* __restrict__ imgF,
                              const int*   __restrict__ kp,
                              float*       __restrict__ out)
{
    __shared__ alignas(16) float sI[DDIM];     // CLS feature row
    __shared__ float sS[LTOK];                 // dot -> sims -> masked sims
    __shared__ float sNt2[LTOK];               // token squared norms
    __shared__ float redv[THREADS];
    __shared__ int   redi[THREADS];
    __shared__ int   topk[MAXK];
    __shared__ float sNi;

    const int b    = blockIdx.x;
    const int t    = threadIdx.x;
    const int lane = t & 31;
    const int wid  = t >> 5;
    const int half = lane >> 4;      // 0: K={0,1}, 1: K={2,3} within each 4-wide K step
    const int m    = lane & 15;      // matrix row within tile

    // ---- load i_feats[b] into LDS, compute ||i|| via block reduction ----
    const float4 fv = reinterpret_cast<const float4*>(iF + (size_t)b * DDIM)[t];
    reinterpret_cast<float4*>(sI)[t] = fv;
    redv[t] = fv.x * fv.x + fv.y * fv.y + fv.z * fv.z + fv.w * fv.w;
    __syncthreads();
    for (int off = THREADS / 2; off > 0; off >>= 1) {
        if (t < off) redv[t] += redv[t + off];
        __syncthreads();
    }
    if (t == 0) sNi = sqrtf(redv[0]);
    __syncthreads();

    // ---- Phase A: per-wave 16-token tiles; fp32 WMMA dot + fused norms ----
    const float* baseImg = imgF + (size_t)b * LTOK * DDIM;
    for (int tl = wid; tl < NTILES; tl += NWAVES) {
        const int row0 = tl * 16;
        // A fragment: lane holds row (row0+m), K offsets half*2 + {0,1}
        const float* pA = baseImg + (size_t)(row0 + m) * DDIM + half * 2;
        // B fragment: broadcast CLS chunk across all 16 N columns (same K mapping)
        const float* pB = sI + half * 2;

        v8f  c  = {0.f, 0.f, 0.f, 0.f, 0.f, 0.f, 0.f, 0.f};
        float sq = 0.f;
        #pragma unroll 4
        for (int d0 = 0; d0 < DDIM; d0 += 4) {
            v2f a  = *reinterpret_cast<const v2f*>(pA + d0);   // global_load_b64
            v2f bb = *reinterpret_cast<const v2f*>(pB + d0);   // ds_load_b64
            // D = A(16x4 f32) * B(4x16 f32) + C ; every N column equals the dot
            c = __builtin_amdgcn_wmma_f32_16x16x4_f32(
                    false, a, false, bb, (short)0, c, false, false);
            sq = __builtin_fmaf(a.x, a.x, sq);
            sq = __builtin_fmaf(a.y, a.y, sq);
        }
        // C layout: VGPR r holds M = r + 8*half (all N columns identical)
        if (m == 0) {
            #pragma unroll
            for (int r = 0; r < 8; ++r) sS[row0 + half * 8 + r] = c[r];
        }
        // lane l and lane l+16 cover the same row's K residues {0,1} / {2,3}
        float sqOther = __shfl_xor(sq, 16, 32);
        if (half == 0) sNt2[row0 + m] = sq + sqOther;
    }
    __syncthreads();

    // ---- Phase B: cosine sims ----
    const float ni = sNi;
    for (int l = t; l < LTOK; l += THREADS) {
        float nt    = sqrtf(sNt2[l]);
        float denom = fmaxf(ni * nt, EPS_);
        sS[l]       = sS[l] / denom;
    }
    __syncthreads();

    // ---- Phase C: k smallest sims (tie -> lowest index, matches lax.top_k(-sims)) ----
    int K = *kp;
    if (K < 1) K = 1;
    if (K > MAXK) K = MAXK;
    for (int sel = 0; sel < K; ++sel) {
        float v = FLT_MAX; int vi = 0x7fffffff;
        for (int l = t; l < LTOK; l += THREADS) {
            float s = sS[l];
            if (s < v) { v = s; vi = l; }
        }
        redv[t] = v; redi[t] = vi;
        __syncthreads();
        for (int off = THREADS / 2; off > 0; off >>= 1) {
            if (t < off) {
                float v2 = redv[t + off]; int i2 = redi[t + off];
                float v1 = redv[t];       int i1 = redi[t];
                if (v2 < v1 || (v2 == v1 && i2 < i1)) { redv[t] = v2; redi[t] = i2; }
            }
            __syncthreads();
        }
        if (t == 0) { int w = redi[0]; topk[sel] = w; sS[w] = FLT_MAX; }
        __syncthreads();
    }

    // ---- Phase D: mean of selected token rows (coalesced float4, L2 hits) ----
    float ax = 0.f, ay = 0.f, az = 0.f, aw = 0.f;
    for (int j = 0; j < K; ++j) {
        const float4 r =
            reinterpret_cast<const float4*>(baseImg + (size_t)topk[j] * DDIM)[t];
        ax += r.x; ay += r.y; az += r.z; aw += r.w;
    }
    const float inv = 1.0f / (float)K;
    float4 o; o.x = ax * inv; o.y = ay * inv; o.z = az * inv; o.w = aw * inv;
    reinterpret_cast<float4*>(out + (size_t)b * DDIM)[t] = o;
}

extern "C" void kernel_launch(void* const* d_in, const int* in_sizes, int n_in,
                              void* d_out, int out_size, void* d_ws, size_t ws_size,
                              hipStream_t stream) {
    const float* iF   = (const float*)d_in[0];   // [256, 1024]
    const float* imgF = (const float*)d_in[1];   // [256, 576, 1024]
    const int*   kp   = (const int*)d_in[2];     // scalar k (=24)
    float*       out  = (float*)d_out;           // [256, 1024]

    dim3 grid(BATCH), block(THREADS);
    hipLaunchKernelGGL(topk_neg_cos_mean_kernel, grid, block, 0, stream,
                       iF, imgF, kp, out);
}